// TTFS_LIF_60052232733276
// MI455X (gfx1250) — compile-verified
//
#include <hip/hip_runtime.h>

// Problem constants (from the reference)
#define TT    512      // T
#define INF   2048     // IN_F
#define OUTF  10       // OUT_F
#define BB    32       // B
#define OPAD  16       // OUT_F padded to WMMA tile width

typedef __attribute__((ext_vector_type(2))) float v2f;
typedef __attribute__((ext_vector_type(8))) float v8f;

// a1 = exp(-1/tau) = exp(-0.5), a2 = exp(-1/tau_s) = exp(-2)
// V0 = 1 / (4^{-1/3} - 4^{-4/3}) = 4^{4/3}/3
#define A1C 0.6065306597126334f
#define A2C 0.1353352832366127f
#define V0C 2.1165347359575992f

// ---------------------------------------------------------------------------
// Kernel 1: y[b][t][o] = sum_n count_t[b][n][t] * W[o][n]   (o padded to 16)
// One wave handles a 16-row t-tile over the FULL K=2048 via WMMA f32 16x16x4.
// ---------------------------------------------------------------------------
__global__ void ttfs_gemm_wmma(const float* __restrict__ count_t,
                               const float* __restrict__ W,
                               float* __restrict__ y /* [B][T][OPAD] */) {
    extern __shared__ float wt[];            // W^T padded: [INF][OPAD] = 128 KB

    const int tid = threadIdx.x;

    // Cooperative stage of W^T (zero-padded columns 10..15) into LDS.
    {
        const int o = tid & 15;
        for (int n = tid >> 4; n < INF; n += 16) {
            wt[n * OPAD + o] = (o < OUTF) ? W[o * INF + n] : 0.0f;
        }
    }
    __syncthreads();

    const int lane = tid & 31;
    const int wave = tid >> 5;
    const int b    = blockIdx.y;
    const int t0   = blockIdx.x * 128 + wave * 16;   // this wave's 16-row t-tile

    // A-matrix (16x4 f32) per-lane source pointer.
    // Lanes 0-15: rows t0+L, K-pair {k0, k0+1}; lanes 16-31: same rows, {k0+2, k0+3}.
    const float* xb = count_t
                    + (size_t)b * INF * TT
                    + (size_t)t0
                    + (lane & 15)
                    + (size_t)(lane >> 4) * 2 * TT;

    // B-matrix (4x16 f32) per-lane LDS index (floats):
    // lanes 0-15 read row k0 (and k0+1 in 2nd VGPR); lanes 16-31 read k0+2 (k0+3).
    const int lb = (lane & 15) + (lane >> 4) * 32;

    v8f c = {0.f, 0.f, 0.f, 0.f, 0.f, 0.f, 0.f, 0.f};

#pragma unroll 8
    for (int i = 0; i < INF / 4; ++i) {       // k0 = 4*i
        v2f a, bm;
        a.x  = xb[(size_t)i * 4 * TT];        // A[m][k0+0] / [k0+2]
        a.y  = xb[(size_t)i * 4 * TT + TT];   // A[m][k0+1] / [k0+3]
        bm.x = wt[i * 64 + lb];               // B[k0+0][n] / [k0+2][n]
        bm.y = wt[i * 64 + lb + 16];          // B[k0+1][n] / [k0+3][n]
        // D = A(16x4) * B(4x16) + C  -> v_wmma_f32_16x16x4_f32
        c = __builtin_amdgcn_wmma_f32_16x16x4_f32(
                /*neg_a=*/false, a, /*neg_b=*/false, bm,
                /*c_mod=*/(short)0, c, /*reuse_a=*/false, /*reuse_b=*/false);
    }

    // C/D layout: VGPR r -> row t0 + r (+8 for upper half-wave); col = lane&15.
    float* yb = y + ((size_t)b * TT + t0 + (lane >> 4) * 8) * OPAD + (lane & 15);
#pragma unroll
    for (int r = 0; r < 8; ++r) {
        yb[r * OPAD] = c[r];
    }
}

// ---------------------------------------------------------------------------
// Kernel 2: dual-exponential scan over t on the tiny projected tensor,
// then v_out = V0*(p1-p2), spike = heaviside(v_out - threshold[t]).
// One thread per (b, o). Outputs concatenated: [v_out | spike], each [B][T][O].
// ---------------------------------------------------------------------------
__global__ void ttfs_scan(const float* __restrict__ y /* [B][T][OPAD] */,
                          const float* __restrict__ threshold /* [T] */,
                          float* __restrict__ out) {
    const int tid = blockIdx.x * blockDim.x + threadIdx.x;
    const int b = tid >> 4;
    const int o = tid & 15;
    if (b >= BB || o >= OUTF) return;

    const float* yb  = y + (size_t)b * TT * OPAD + o;
    float* vout      = out + (size_t)b * TT * OUTF + o;
    float* sout      = vout + (size_t)BB * TT * OUTF;

    float p1 = 0.0f, p2 = 0.0f;
#pragma unroll 4
    for (int t = 0; t < TT; ++t) {
        const float yt = yb[(size_t)t * OPAD];
        p1 = A1C * (yt + p1);
        p2 = A2C * (yt + p2);
        const float v = V0C * (p1 - p2);
        vout[(size_t)t * OUTF] = v;
        sout[(size_t)t * OUTF] = (v >= threshold[t]) ? 1.0f : 0.0f;
    }
}

// ---------------------------------------------------------------------------
extern "C" void kernel_launch(void* const* d_in, const int* in_sizes, int n_in,
                              void* d_out, int out_size, void* d_ws, size_t ws_size,
                              hipStream_t stream) {
    const float* threshold = (const float*)d_in[0];   // [T]
    const float* count_t   = (const float*)d_in[1];   // [B, IN_F, T]
    const float* W         = (const float*)d_in[2];   // [OUT_F, IN_F]
    float* out = (float*)d_out;                       // [v_out | spike], 2*B*T*O
    float* y   = (float*)d_ws;                        // [B][T][OPAD] = 1 MiB

    // GEMM: grid (T/128, B), 8 waves/block, each wave a 16-row t-tile, full K.
    dim3 grid1(TT / 128, BB);
    ttfs_gemm_wmma<<<grid1, 256, INF * OPAD * sizeof(float), stream>>>(count_t, W, y);

    // Scan: one thread per (b, o) pair; B*16 = 512 threads.
    ttfs_scan<<<(BB * OPAD + 255) / 256, 256, 0, stream>>>(y, threshold, out);
}